// PaiConv_15702400434772
// MI455X (gfx1250) — compile-verified
//
#include <hip/hip_runtime.h>

typedef __attribute__((ext_vector_type(2))) float v2f;
typedef __attribute__((ext_vector_type(8))) float v8f;

#define B_      8
#define C_      64
#define N_      4096
#define K_      20
#define M_      8
#define OUTC_   64
#define CM_     (C_ * M_)          /* 512 */
#define NEG_SLOPE 0.2f

#define TILE_PTS   16
#define AGG_STRIDE 516             /* 512 + 4 pad: 16 A-rows hit distinct LDS banks */

// ---------- kernel 1: feature (B,C,N) -> ftT (B,N,C), LDS-tiled transpose ----------
__global__ __launch_bounds__(256) void k_transpose_feature(const float* __restrict__ f,
                                                           float* __restrict__ ftT) {
    __shared__ float tile[64][65];
    const int blk = blockIdx.x;
    const int b   = blk >> 6;              // 64 n-tiles per batch
    const int n0  = (blk & 63) << 6;
    const int tid = threadIdx.x;
#pragma unroll
    for (int i = 0; i < 16; ++i) {
        int e = tid + i * 256;
        int c = e >> 6, n = e & 63;        // consecutive tid -> consecutive n (coalesced read)
        tile[c][n] = f[(size_t)(b * C_ + c) * N_ + n0 + n];
    }
    __syncthreads();
#pragma unroll
    for (int i = 0; i < 16; ++i) {
        int e = tid + i * 256;
        int n = e >> 6, c = e & 63;        // consecutive tid -> consecutive c (coalesced write)
        ftT[((size_t)(b * N_ + n0 + n)) * C_ + c] = tile[c][n];
    }
}

// ---------- kernel 2: conv_w (64,512) -> wT (512,64) ----------
__global__ __launch_bounds__(256) void k_transpose_w(const float* __restrict__ w,
                                                     float* __restrict__ wT) {
    int e  = blockIdx.x * 256 + threadIdx.x;   // 32768 elements total
    int oc = e >> 9, kk = e & 511;
    wT[kk * OUTC_ + oc] = w[e];
}

// ---------- kernel 3: fused PaiConv ----------
__global__ __launch_bounds__(128) void k_paiconv(
    const float* __restrict__ x, const float* __restrict__ feature,
    const long long* __restrict__ nidx, const float* __restrict__ kernels,
    const float* __restrict__ conv_b,
    const float* __restrict__ ftT, const float* __restrict__ wT,
    float* __restrict__ out)
{
    __shared__ float kerLds[3][M_];
    __shared__ float perm[TILE_PTS][K_][M_];
    __shared__ int   idxLds[TILE_PTS][K_];
    __shared__ float agg[TILE_PTS][AGG_STRIDE];

    const int tid = threadIdx.x;
    const int blk = blockIdx.x;
    const int b   = blk >> 8;                  // 256 point-tiles per batch
    const int n0  = (blk & 255) * TILE_PTS;

    if (tid < 3 * M_) kerLds[tid / M_][tid % M_] = kernels[tid];
    __syncthreads();

    // ---- Stage A: permatrix = x_rel @ kernels + one_pad ----
    for (int t = tid; t < TILE_PTS * K_; t += 128) {
        const int p = t / K_, k = t % K_;
        const long long* row = nidx + (size_t)(b * N_ + n0 + p) * K_;
        const int j  = (int)row[k];
        const int j0 = (int)row[0];
        idxLds[p][k] = j;
        const float rx = x[(b * 3 + 0) * N_ + j] - x[(b * 3 + 0) * N_ + j0];
        const float ry = x[(b * 3 + 1) * N_ + j] - x[(b * 3 + 1) * N_ + j0];
        const float rz = x[(b * 3 + 2) * N_ + j] - x[(b * 3 + 2) * N_ + j0];
#pragma unroll
        for (int m = 0; m < M_; ++m) {
            float v = rx * kerLds[0][m] + ry * kerLds[1][m] + rz * kerLds[2][m];
            if (k == 0 && m == 0) v += 1.0f;   // one_pad
            perm[p][k][m] = v;
        }
    }
    __syncthreads();

    // ---- softmax over k, one (point, m) column per thread (exactly 128 tasks) ----
    {
        const int p = tid >> 3, m = tid & 7;
        float mx = -1e30f;
        for (int k = 0; k < K_; ++k) mx = fmaxf(mx, perm[p][k][m]);
        float e[K_], s = 0.0f;
        for (int k = 0; k < K_; ++k) { e[k] = expf(perm[p][k][m] - mx); s += e[k]; }
        const float inv = 1.0f / s;
        for (int k = 0; k < K_; ++k) perm[p][k][m] = e[k] * inv;
    }
    __syncthreads();

    // ---- Stage B: agg[p][c*8+m] = sum_k ftT[b, idx[p][k], c] * perm[p][k][m] ----
    // 64 lanes share one point -> ftT gathers are 256B-contiguous per (p,k).
    for (int it = 0; it < 8; ++it) {
        const int p = (tid >> 6) + it * 2;
        const int c = tid & 63;
        const float* fb = ftT + (size_t)b * N_ * C_;
        float acc[M_];
#pragma unroll
        for (int m = 0; m < M_; ++m) acc[m] = 0.0f;
        for (int k = 0; k < K_; ++k) {
            const int j = idxLds[p][k];
            const float fv = fb[(size_t)j * C_ + c];
#pragma unroll
            for (int m = 0; m < M_; ++m) acc[m] = fmaf(fv, perm[p][k][m], acc[m]);
        }
#pragma unroll
        for (int m = 0; m < M_; ++m) agg[p][c * M_ + m] = acc[m];
    }
    __syncthreads();

    // ---- Stage C: (16 x 512) @ wT(512 x 64) via V_WMMA_F32_16X16X4_F32 ----
    // wave w -> oc tile [16w, 16w+16). A: lane<16 holds (M=lane, K0/K1),
    // lane>=16 holds (M=lane-16, K2/K3); B symmetric on K rows.
    const int wave   = tid >> 5;
    const int lane   = tid & 31;
    const int row    = lane & 15;
    const int khi    = lane >> 4;
    const int ocBase = wave * 16;
    const int oc     = ocBase + row;

    v8f d = {};
    const float* aggRow = &agg[row][0];
    for (int k0 = 0; k0 < CM_; k0 += 4) {
        const int ka = k0 + 2 * khi;
        v2f a, bb;
        a.x  = aggRow[ka];
        a.y  = aggRow[ka + 1];
        bb.x = wT[ka * OUTC_ + oc];
        bb.y = wT[(ka + 1) * OUTC_ + oc];
        d = __builtin_amdgcn_wmma_f32_16x16x4_f32(false, a, false, bb,
                                                  (short)0, d, false, false);
    }

    // D layout: VGPR r -> point (r + 8*khi), column oc. 8 consecutive-n stores/lane.
    const float bias = conv_b[oc];
    const size_t base = ((size_t)(b * OUTC_ + oc)) * N_ + n0 + khi * 8;
#pragma unroll
    for (int r = 0; r < 8; ++r) {
        float v = d[r] + bias;
        v = (v > 0.0f) ? v : NEG_SLOPE * v;
        out[base + r] = v + feature[base + r];
    }
}

extern "C" void kernel_launch(void* const* d_in, const int* in_sizes, int n_in,
                              void* d_out, int out_size, void* d_ws, size_t ws_size,
                              hipStream_t stream) {
    (void)in_sizes; (void)n_in; (void)out_size; (void)ws_size;
    const float*     x        = (const float*)d_in[0];
    const float*     feature  = (const float*)d_in[1];
    const long long* nidx     = (const long long*)d_in[2];
    const float*     kernels  = (const float*)d_in[3];
    const float*     conv_w   = (const float*)d_in[4];
    const float*     conv_b   = (const float*)d_in[5];
    float*           out      = (float*)d_out;

    float* ftT = (float*)d_ws;                     // B*N*C floats (8 MB)
    float* wT  = ftT + (size_t)B_ * N_ * C_;       // 512*64 floats (128 KB)

    k_transpose_feature<<<B_ * (N_ / 64), 256, 0, stream>>>(feature, ftT);
    k_transpose_w<<<(CM_ * OUTC_) / 256, 256, 0, stream>>>(conv_w, wT);
    k_paiconv<<<B_ * (N_ / TILE_PTS), 128, 0, stream>>>(x, feature, nidx, kernels,
                                                        conv_b, ftT, wT, out);
}